// Patcher_1529008357476
// MI455X (gfx1250) — compile-verified
//
#include <hip/hip_runtime.h>

typedef float v2f __attribute__((ext_vector_type(2)));
typedef float v8f __attribute__((ext_vector_type(8)));

#define NPTS   8192
#define NDIM   3
#define KNN    16
#define TILE   16
#define NTILES (NPTS / TILE)        // 512
#define WAVES  8
#define THREADS (WAVES * 32)
#define QTILES_PER_BLOCK WAVES
#define IDX_MASK 0x1FFFLL           // 13 bits: NPTS = 8192

// LDS: 8192 * float4 (x,y,z,|p|^2) = 128 KB  +  8 waves * 16 q * 32 keys * 8B = 32 KB
#define SMEM_BYTES (NPTS * sizeof(float4) + (size_t)WAVES * TILE * 2 * KNN * sizeof(double))

// Raw f64 min/max: bypass the backend's canonicalize (v_max x,x,x) insertion —
// our packed keys are always finite, so IEEE minNum/maxNum == plain min/max.
__device__ __forceinline__ double dmin(double a, double b) {
    double r;
    asm("v_min_num_f64 %0, %1, %2" : "=v"(r) : "v"(a), "v"(b));
    return r;
}
__device__ __forceinline__ double dmax(double a, double b) {
    double r;
    asm("v_max_num_f64 %0, %1, %2" : "=v"(r) : "v"(a), "v"(b));
    return r;
}

__global__ __launch_bounds__(THREADS)
void knn_group_kernel(const float* __restrict__ p, float* __restrict__ out) {
    extern __shared__ float4 pts[];                  // [NPTS] {x,y,z,sq}
    double* mergebuf = (double*)(pts + NPTS);        // [WAVES][TILE][2][KNN] packed keys

    const int b   = blockIdx.y;
    const int tid = threadIdx.x;
    const float* pb = p + (size_t)b * NDIM * NPTS;

    // ---- stage this batch's points into LDS as {x,y,z,|p|^2} ----
    for (int n = tid; n < NPTS; n += THREADS) {
        float x = pb[n];
        float y = pb[NPTS + n];
        float z = pb[2 * NPTS + n];
        pts[n] = make_float4(x, y, z, x * x + y * y + z * z);
    }
    __syncthreads();

    const int wave  = tid >> 5;          // wave32
    const int lane  = tid & 31;
    const int l16   = lane & 15;
    const int lhalf = lane >> 4;         // 0: lanes 0-15, 1: lanes 16-31

    const int qt = blockIdx.x * QTILES_PER_BLOCK + wave;   // query tile 0..511
    const int qg = qt * TILE + l16;                        // this lane's query point

    // B operand = Q^T (4x16, K=3 padded to 4).
    // B layout (32-bit, 4x16): lanes 0-15 hold K=0,1 for col N=lane; lanes 16-31 hold K=2,3.
    float4 qp = pts[qg];
    v2f Bop;
    Bop[0] = lhalf ? qp.z : qp.x;
    Bop[1] = lhalf ? 0.0f : qp.y;
    const float qsq = qp.w;

    // register-resident sorted (ascending) top-16 of packed f64 keys:
    //   key = (double)d2  with the 13-bit candidate index OR'ed into the low
    //   mantissa bits (zero after f32->f64): orders by distance, ties by index.
    double bk[KNN];
#pragma unroll
    for (int s = 0; s < KNN; ++s) bk[s] = __builtin_inf();

    for (int ct = 0; ct < NTILES; ++ct) {
        const int cbase  = ct * TILE;
        const int mybase = cbase + 8 * lhalf;   // first of this lane's 8 candidate rows

        // A operand = -2 * candidate tile (16x4, K padded).
        // A layout (32-bit, 16x4): lanes 0-15 hold K=0,1 for row M=lane; lanes 16-31 hold K=2,3.
        float4 cp = pts[cbase + l16];
        v2f Aop;
        Aop[0] = -2.0f * (lhalf ? cp.z : cp.x);
        Aop[1] = -2.0f * (lhalf ? 0.0f : cp.y);

        // C operand: csq[m] + qsq  (C/D layout: VGPR r -> row M = r + 8*lhalf, col N = l16)
        v8f Cin;
#pragma unroll
        for (int r = 0; r < 8; ++r) Cin[r] = pts[mybase + r].w + qsq;

        // d2[m][n] = qsq + csq[m] - 2 * (c_m . q_n), exact f32 via WMMA
        v8f d2 = __builtin_amdgcn_wmma_f32_16x16x4_f32(
            /*neg_a=*/false, Aop, /*neg_b=*/false, Bop,
            /*c_mod=*/(short)0, Cin, /*reuse_a=*/false, /*reuse_b=*/false);

        // Branchless sorted insert via the min/max identity:
        //   new[s] = min(max(key, a[s-1]), a[s]) ; new[0] = min(key, a[0])
        // 31 raw f64 min/max ops, guarded by one compare + execz branch.
#pragma unroll
        for (int r = 0; r < 8; ++r) {
            double key = __longlong_as_double(
                __double_as_longlong((double)d2[r]) | (long long)(mybase + r));
            if (key < bk[KNN - 1]) {        // v_cmpx + s_cbranch_execz, usually skipped
#pragma unroll
                for (int s = KNN - 1; s >= 1; --s)
                    bk[s] = dmin(dmax(key, bk[s - 1]), bk[s]);
                bk[0] = dmin(key, bk[0]);
            }
        }
    }

    // ---- dump both sorted half-lists for this query ----
    double* mylist = mergebuf + ((size_t)(wave * TILE + l16) * 2 + lhalf) * KNN;
#pragma unroll
    for (int s = 0; s < KNN; ++s) mylist[s] = bk[s];
    __syncthreads();

    // ---- merge the two sorted 16-lists (packed keys compare directly),
    //      gather coords from LDS, write [B][3][N][16] output ----
    if (lhalf == 0) {
        const double* la = mergebuf + ((size_t)(wave * TILE + l16) * 2) * KNN;
        const double* lb = la + KNN;
        float* outx = out + ((size_t)(b * NDIM + 0) * NPTS + qg) * KNN;
        float* outy = out + ((size_t)(b * NDIM + 1) * NPTS + qg) * KNN;
        float* outz = out + ((size_t)(b * NDIM + 2) * NPTS + qg) * KNN;
        int ia = 0, ib = 0;
        for (int kk = 0; kk < KNN; ++kk) {
            double a  = la[ia];
            double bb = lb[ib];
            bool takeA = a < bb;
            double w = takeA ? a : bb;
            ia += takeA ? 1 : 0;
            ib += takeA ? 0 : 1;
            int idx = (int)(__double_as_longlong(w) & IDX_MASK);
            float4 pt = pts[idx];
            outx[kk] = pt.x;
            outy[kk] = pt.y;
            outz[kk] = pt.z;
        }
    }
}

extern "C" void kernel_launch(void* const* d_in, const int* in_sizes, int n_in,
                              void* d_out, int out_size, void* d_ws, size_t ws_size,
                              hipStream_t stream) {
    const float* p = (const float*)d_in[0];
    float* out = (float*)d_out;
    const int B = in_sizes[0] / (NDIM * NPTS);   // = 4

    (void)hipFuncSetAttribute((const void*)knn_group_kernel,
                              hipFuncAttributeMaxDynamicSharedMemorySize,
                              (int)SMEM_BYTES);

    dim3 grid(NTILES / QTILES_PER_BLOCK, B);
    knn_group_kernel<<<grid, THREADS, SMEM_BYTES, stream>>>(p, out);
}